// ResidualElementDependentInteractionBlock_343597384377
// MI455X (gfx1250) — compile-verified
//
#include <hip/hip_runtime.h>

namespace {

constexpr int NN = 10000;      // nodes
constexpr int NE = 160000;     // edges
constexpr float INV_SQRT3F = 0.5773502691896258f;
constexpr float INV_AVG = 1.0f / 16.0f;   // AVG_NEIGH = 16.0

typedef float v2f __attribute__((ext_vector_type(2)));
typedef float v8f __attribute__((ext_vector_type(8)));

__device__ __forceinline__ v8f wmma4(v2f a, v2f b, v8f c) {
  // D = A(16x4) * B(4x16) + C, fp32 WMMA
  return __builtin_amdgcn_wmma_f32_16x16x4_f32(false, a, false, b, (short)0, c,
                                               false, false);
}

__device__ __forceinline__ v8f z8() {
  v8f z = {0.f, 0.f, 0.f, 0.f, 0.f, 0.f, 0.f, 0.f};
  return z;
}

// Async global->LDS 16B copy (GVS mode: saddr + 32-bit vgpr byte offset).
// vdst carries the LDS byte address; tracked by ASYNCcnt.
__device__ __forceinline__ void async_g2l_b128(const void* sbase, unsigned gvoff,
                                               unsigned ldsaddr) {
  asm volatile("global_load_async_to_lds_b128 %0, %1, %2"
               :
               : "v"(ldsaddr), "v"(gvoff), "s"(sbase)
               : "memory");
}

__device__ __forceinline__ void wait_async0() {
#if __has_builtin(__builtin_amdgcn_s_wait_asynccnt)
  __builtin_amdgcn_s_wait_asynccnt(0);
#else
  asm volatile("s_wait_asynccnt 0x0" ::: "memory");
#endif
}

// LDS byte offset of a shared-memory pointer (flat addr low 32 bits).
__device__ __forceinline__ unsigned lds_addr(const void* p) {
  return (unsigned)(uintptr_t)p;
}

// ---------------------------------------------------------------- zero scratch
__global__ __launch_bounds__(256) void zero_f32(float* __restrict__ p, int n4) {
  int i = blockIdx.x * blockDim.x + threadIdx.x;
  float4 z = make_float4(0.f, 0.f, 0.f, 0.f);
  for (; i < n4; i += gridDim.x * blockDim.x)
    reinterpret_cast<float4*>(p)[i] = z;
}

// ------------------------------------------------- node: up-proj + skip (sc)
// block = 16 nodes, 8 waves; wave w owns output columns [16w, 16w+16)
// h0 = x0 @ Wup0 ; h1[m] = x1[:,:,m] @ Wup1
// sc0 = (x0 (x) attrs) @ Wsk0 (K=1280) ; sc1[m] likewise with Wsk1
// sc written directly into final out layout.
__global__ __launch_bounds__(256) void node_kernel(
    const float* __restrict__ attrs, const float* __restrict__ feats,
    const float* __restrict__ Wup0, const float* __restrict__ Wup1,
    const float* __restrict__ Wsk0, const float* __restrict__ Wsk1,
    float* __restrict__ h0, float* __restrict__ h1, float* __restrict__ out) {
  __shared__ float xs[16 * 516];   // node_feats tile; 516*4 = 2064B rows: 16B
                                   // aligned, stride%64banks==4 -> conflict-free
  __shared__ float as_[16 * 12];   // node_attrs tile
  const int tid = threadIdx.x;
  const int base = blockIdx.x * 16;

  // async global->LDS stage of the first 128 floats of the 16 feature rows (512 x b128)
  {
    const float* gsrc = feats + (size_t)base * 512;
    const unsigned xs0 = lds_addr(&xs[0]);
#pragma unroll
    for (int rep = 0; rep < 2; ++rep) {
      const int idx = rep * 256 + tid;
      const int row = idx >> 5, c4 = idx & 31;  // 32 x 16B per row
      async_g2l_b128(gsrc, (unsigned)((row * 512 + c4 * 4) * 4),
                     xs0 + (unsigned)((row * 516 + c4 * 4) * 4));
    }
  }
  if (tid < 160) {
    const int row = tid / 10, a = tid - row * 10;
    as_[row * 12 + a] = attrs[(base + row) * 10 + a];
  }
  wait_async0();
  __syncthreads();

  const int wv = tid >> 5, lane = tid & 31, hf = lane >> 4, r = lane & 15;
  const int col = wv * 16 + r;
  const float* xrow = &xs[r * 516];
  const float* arow = &as_[r * 12];

  v8f ah0 = z8();
  v8f ah1[3] = {z8(), z8(), z8()};
  // ---- up projections, K = 128
  for (int k0 = 0; k0 < 128; k0 += 4) {
    const int kk = k0 + 2 * hf;
    v2f a; a.x = xrow[kk]; a.y = xrow[kk + 1];
    v2f b; b.x = Wup0[kk * 128 + col]; b.y = Wup0[(kk + 1) * 128 + col];
    ah0 = wmma4(a, b, ah0);
    v2f b1; b1.x = Wup1[kk * 128 + col]; b1.y = Wup1[(kk + 1) * 128 + col];
#pragma unroll
    for (int m = 0; m < 3; ++m) {
      v2f a1;
      a1.x = xrow[128 + kk * 3 + m];
      a1.y = xrow[128 + (kk + 1) * 3 + m];
      ah1[m] = wmma4(a1, b1, ah1[m]);
    }
  }

  v8f as0 = z8();
  v8f as1[3] = {z8(), z8(), z8()};
  // ---- element-dependent skip, K = 1280, composite k = u*10 + a
  int u = 0, ai = 2 * hf;  // tracks (k0 + 2*hf) as (u, ai)
  for (int k0 = 0; k0 < 1280; k0 += 4) {
    const int kk = k0 + 2 * hf;
    const int u0 = u, a0 = ai;
    int u1 = u, a1 = ai + 1;
    if (a1 >= 10) { a1 -= 10; ++u1; }
    const float t0 = arow[a0], t1 = arow[a1];
    v2f b0; b0.x = Wsk0[kk * 128 + col]; b0.y = Wsk0[(kk + 1) * 128 + col];
    v2f b1; b1.x = Wsk1[kk * 128 + col]; b1.y = Wsk1[(kk + 1) * 128 + col];
    v2f q; q.x = xrow[u0] * t0; q.y = xrow[u1] * t1;
    as0 = wmma4(q, b0, as0);
#pragma unroll
    for (int m = 0; m < 3; ++m) {
      v2f q1;
      q1.x = xrow[128 + u0 * 3 + m] * t0;
      q1.y = xrow[128 + u1 * 3 + m] * t1;
      as1[m] = wmma4(q1, b1, as1[m]);
    }
    ai += 4;
    if (ai >= 10) { ai -= 10; ++u; }
  }

#pragma unroll
  for (int i = 0; i < 8; ++i) {
    const int n = base + i + 8 * hf;
    h0[(size_t)n * 128 + col] = ah0[i];
#pragma unroll
    for (int m = 0; m < 3; ++m)
      h1[(size_t)m * ((size_t)NN * 128) + (size_t)n * 128 + col] = ah1[m][i];
    out[(size_t)n * 512 + col] = as0[i];                       // sc0
#pragma unroll
    for (int m = 0; m < 3; ++m)
      out[(size_t)n * 512 + 128 + col * 3 + m] = as1[m][i];    // sc1 (k*3+m)
  }
}

// --------------------------- edges: fused tpw GEMM + messages + atomic scatter
// block = 16 edges, 8 waves; K = 80 (a*8+e), N = 512.  tpw kept in LDS only.
__global__ __launch_bounds__(256) void edge_kernel(
    const float* __restrict__ attrs, const float* __restrict__ eattr,
    const float* __restrict__ efeat, const int* __restrict__ eidx,
    const float* __restrict__ Wt, const float* __restrict__ h0,
    const float* __restrict__ h1, float* __restrict__ msg0,
    float* __restrict__ msg1) {
  __shared__ float q_s[16 * 84];    // A matrix: outer(attrs[snd], feats)
  __shared__ float tp_s[16 * 516];  // tpw tile (w1|w2|w3|w4)
  __shared__ float y_s[16 * 4];
  __shared__ int snd_s[16], rcv_s[16];
  const int tid = threadIdx.x;
  const int ebase = blockIdx.x * 16;

  if (tid < 16) {
    const int e = ebase + tid;
    snd_s[tid] = eidx[e];
    rcv_s[tid] = eidx[NE + e];
#pragma unroll
    for (int m = 0; m < 4; ++m) y_s[tid * 4 + m] = eattr[e * 4 + m];
  }
  __syncthreads();
  for (int i = tid; i < 16 * 80; i += 256) {
    const int e = i / 80, kk = i - e * 80;
    const int a = kk >> 3, f = kk & 7;
    q_s[e * 84 + kk] = attrs[snd_s[e] * 10 + a] * efeat[(ebase + e) * 8 + f];
  }
  __syncthreads();

  const int wv = tid >> 5, lane = tid & 31, hf = lane >> 4, r = lane & 15;
  v8f acc[4] = {z8(), z8(), z8(), z8()};
  const float* qrow = &q_s[r * 84];
  for (int k0 = 0; k0 < 80; k0 += 4) {
    const int kk = k0 + 2 * hf;
    v2f a; a.x = qrow[kk]; a.y = qrow[kk + 1];
#pragma unroll
    for (int jj = 0; jj < 4; ++jj) {
      const int col = (wv * 4 + jj) * 16 + r;
      v2f b; b.x = Wt[kk * 512 + col]; b.y = Wt[(kk + 1) * 512 + col];
      acc[jj] = wmma4(a, b, acc[jj]);
    }
  }
#pragma unroll
  for (int jj = 0; jj < 4; ++jj) {
    const int col = (wv * 4 + jj) * 16 + r;
#pragma unroll
    for (int i = 0; i < 8; ++i) tp_s[(i + 8 * hf) * 516 + col] = acc[jj][i];
  }
  __syncthreads();

  // message + scatter: 16 threads per edge, each covers u = (tid&15)+16*i
  const int el = tid >> 4;
  const int s = snd_s[el], rc = rcv_s[el];
  const float Y0 = y_s[el * 4 + 0];
  const float Y10 = y_s[el * 4 + 1], Y11 = y_s[el * 4 + 2],
              Y12 = y_s[el * 4 + 3];
  const float* tp = &tp_s[el * 516];
  const size_t so = (size_t)s * 128;
  const size_t ro = (size_t)rc * 256;
  const size_t PL = (size_t)NN * 128;
  const size_t MP = (size_t)NN * 256;
#pragma unroll
  for (int i = 0; i < 8; ++i) {
    const int uu = (tid & 15) + 16 * i;
    const float w1 = tp[uu], w2 = tp[128 + uu], w3 = tp[256 + uu],
                w4 = tp[384 + uu];
    const float hv = h0[so + uu];
    const float g0 = h1[so + uu];
    const float g1 = h1[PL + so + uu];
    const float g2 = h1[2 * PL + so + uu];
    const float dot = g0 * Y10 + g1 * Y11 + g2 * Y12;
    atomicAdd(&msg0[ro + uu], w1 * hv * Y0);
    atomicAdd(&msg0[ro + 128 + uu], w4 * (INV_SQRT3F * dot));
    const float w2h = w2 * hv, w3y = w3 * Y0;
    atomicAdd(&msg1[ro + uu], w2h * Y10);
    atomicAdd(&msg1[MP + ro + uu], w2h * Y11);
    atomicAdd(&msg1[2 * MP + ro + uu], w2h * Y12);
    atomicAdd(&msg1[ro + 128 + uu], g0 * w3y);
    atomicAdd(&msg1[MP + ro + 128 + uu], g1 * w3y);
    atomicAdd(&msg1[2 * MP + ro + 128 + uu], g2 * w3y);
  }
}

// ----------------------------------------------- output linear + accumulate sc
// msg1 tiles staged once per block in LDS via async loads (was: every wave
// re-reading the same tile from global).
__global__ __launch_bounds__(256) void out_kernel(
    const float* __restrict__ msg0, const float* __restrict__ msg1,
    const float* __restrict__ Wl0, const float* __restrict__ Wl1,
    float* __restrict__ out) {
  __shared__ float m1s[3 * 16 * 260];  // 260*4 = 1040B rows: 16B aligned,
                                       // stride%64banks==4 -> conflict-free
  const int tid = threadIdx.x;
  const int base = blockIdx.x * 16;
  const size_t MP = (size_t)NN * 256;

  {
    const unsigned m10 = lds_addr(&m1s[0]);
#pragma unroll
    for (int rep = 0; rep < 3; ++rep) {  // 768 x b128 = 3 planes x 16 rows x 64 floats
      const int idx = rep * 256 + tid;
      const int pl = idx >> 8, rem = idx & 255;
      const int row = rem >> 4, c4 = rem & 15;
      async_g2l_b128(
          msg1,
          (unsigned)(((size_t)pl * MP + (size_t)(base + row) * 256 + c4 * 4) *
                     4),
          m10 + (unsigned)((pl * (16 * 260) + row * 260 + c4 * 4) * 4));
    }
  }
  wait_async0();
  __syncthreads();

  const int wv = tid >> 5, lane = tid & 31, hf = lane >> 4, r = lane & 15;
  const int col = wv * 16 + r;
  const float* a0 = msg0 + (size_t)(base + r) * 256;
  const float* a1l = &m1s[r * 260];
  v8f acc0 = z8();
  v8f acc1[3] = {z8(), z8(), z8()};
  for (int k0 = 0; k0 < 256; k0 += 4) {
    const int kk = k0 + 2 * hf;
    v2f a; a.x = a0[kk]; a.y = a0[kk + 1];
    v2f b; b.x = Wl0[kk * 128 + col]; b.y = Wl0[(kk + 1) * 128 + col];
    acc0 = wmma4(a, b, acc0);
    v2f b1; b1.x = Wl1[kk * 128 + col]; b1.y = Wl1[(kk + 1) * 128 + col];
#pragma unroll
    for (int m = 0; m < 3; ++m) {
      v2f am;
      am.x = a1l[m * (16 * 260) + kk];
      am.y = a1l[m * (16 * 260) + kk + 1];
      acc1[m] = wmma4(am, b1, acc1[m]);
    }
  }
#pragma unroll
  for (int i = 0; i < 8; ++i) {
    const int n = base + i + 8 * hf;
    const size_t o = (size_t)n * 512;
    out[o + col] += acc0[i] * INV_AVG;
#pragma unroll
    for (int m = 0; m < 3; ++m)
      out[o + 128 + col * 3 + m] += acc1[m][i] * INV_AVG;
  }
}

}  // namespace

extern "C" void kernel_launch(void* const* d_in, const int* in_sizes, int n_in,
                              void* d_out, int out_size, void* d_ws,
                              size_t ws_size, hipStream_t stream) {
  const float* node_attrs = (const float*)d_in[0];
  const float* node_feats = (const float*)d_in[1];
  const float* edge_attrs = (const float*)d_in[2];
  const float* edge_feats = (const float*)d_in[3];
  const int* edge_index = (const int*)d_in[4];
  const float* W_up0 = (const float*)d_in[5];
  const float* W_up1 = (const float*)d_in[6];
  const float* W_tpw = (const float*)d_in[7];
  const float* W_lin0 = (const float*)d_in[8];
  const float* W_lin1 = (const float*)d_in[9];
  const float* W_sk0 = (const float*)d_in[10];
  const float* W_sk1 = (const float*)d_in[11];
  float* out = (float*)d_out;

  // workspace layout (floats): h0 | h1(3 planes) | msg0 | msg1(3 planes)
  float* ws = (float*)d_ws;
  float* h0 = ws;                              // NN*128
  float* h1 = h0 + (size_t)NN * 128;           // 3*NN*128
  float* msg0 = h1 + (size_t)3 * NN * 128;     // NN*256
  float* msg1 = msg0 + (size_t)NN * 256;       // 3*NN*256

  const int msg_f4 = (NN * 1024) / 4;          // msg0+msg1 contiguous, in float4
  zero_f32<<<2048, 256, 0, stream>>>(msg0, msg_f4);
  node_kernel<<<NN / 16, 256, 0, stream>>>(node_attrs, node_feats, W_up0, W_up1,
                                           W_sk0, W_sk1, h0, h1, out);
  edge_kernel<<<NE / 16, 256, 0, stream>>>(node_attrs, edge_attrs, edge_feats,
                                           edge_index, W_tpw, h0, h1, msg0,
                                           msg1);
  out_kernel<<<NN / 16, 256, 0, stream>>>(msg0, msg1, W_lin0, W_lin1, out);
}